// Encoder_12111807774759
// MI455X (gfx1250) — compile-verified
//
#include <hip/hip_runtime.h>
#include <math.h>

typedef float v2f __attribute__((ext_vector_type(2)));
typedef float v8f __attribute__((ext_vector_type(8)));

#define K_B 128
#define K_L 8
#define K_E 384
#define K_R 500
#define K_HP 4
#define K_HM 6
#define K_LOSS 9   // 7 step losses + 2 entropy columns
#define K_PRED (K_R + 2)  // 502

// ---------------- device helpers (wave32) ----------------
__device__ __forceinline__ float wave_sum(float v) {
  for (int o = 16; o > 0; o >>= 1) v += __shfl_xor(v, o, 32);
  return v;
}
__device__ __forceinline__ float wave_max(float v) {
  for (int o = 16; o > 0; o >>= 1) v = fmaxf(v, __shfl_xor(v, o, 32));
  return v;
}
__device__ __forceinline__ int imin(int a, int b) { return a < b ? a : b; }

// ---------------- fp32 WMMA GEMM ----------------
// C[M,N](ldc) = alpha * (A[M,K](lda) x B) + bias[N]; accum!=0 -> C += result.
// BT=1: B stored [N,K](ldb), multiply by B^T (the x @ W.T pattern).
// BT=0: B stored [K,N](ldb).
// Requires K % 4 == 0 and even lda/ldb (all shapes here satisfy this).
// Each wave computes a 16x32 output block (2 N-tiles) so one A fragment feeds
// two v_wmma_f32_16x16x4_f32 ops. Edge handling: indices are CLAMPED for the
// loads (out-of-range A rows / B cols only pollute unstored D rows/cols), and
// stores are guarded -> the inner loop is completely branch-free.
template<bool BT>
__global__ void gemm_wmma(float* C, const float* __restrict__ A,
                          const float* __restrict__ Bm, const float* __restrict__ bias,
                          int M, int N, int K, int lda, int ldb, int ldc,
                          float alpha, int accum)
{
  int wave = (blockIdx.x * blockDim.x + threadIdx.x) >> 5;
  int ntiles = (N + 15) >> 4;
  int npair  = (ntiles + 1) >> 1;
  int mtiles = (M + 15) >> 4;
  if (wave >= npair * mtiles) return;           // wave-uniform exit
  int lane = threadIdx.x & 31;
  int tm = wave / npair, tp = wave % npair;
  int half = lane >> 4, lc = lane & 15;
  int arow  = tm * 16 + lc;                     // A: M = lane%16
  int bcol0 = tp * 32 + lc;                     // B tile 0: N = lane%16
  int bcol1 = bcol0 + 16;                       // B tile 1
  int arowc  = imin(arow,  M - 1);              // clamped for loads
  int bc0    = imin(bcol0, N - 1);
  int bc1    = imin(bcol1, N - 1);
  const float* ap = A + (size_t)arowc * lda + half * 2;
  v8f acc0 = {0.f, 0.f, 0.f, 0.f, 0.f, 0.f, 0.f, 0.f};
  v8f acc1 = {0.f, 0.f, 0.f, 0.f, 0.f, 0.f, 0.f, 0.f};
  if (BT) {
    const float* bp0 = Bm + (size_t)bc0 * ldb + half * 2;
    const float* bp1 = Bm + (size_t)bc1 * ldb + half * 2;
#pragma unroll 4
    for (int k0 = 0; k0 < K; k0 += 4) {
      v2f a  = *(const v2f*)(ap  + k0);
      v2f b0 = *(const v2f*)(bp0 + k0);
      v2f b1 = *(const v2f*)(bp1 + k0);
      acc0 = __builtin_amdgcn_wmma_f32_16x16x4_f32(false, a, false, b0, (short)0, acc0, false, false);
      acc1 = __builtin_amdgcn_wmma_f32_16x16x4_f32(false, a, false, b1, (short)0, acc1, false, false);
    }
  } else {
    const float* bp = Bm + (size_t)(half * 2) * ldb;
#pragma unroll 4
    for (int k0 = 0; k0 < K; k0 += 4) {
      v2f a = *(const v2f*)(ap + k0);
      const float* bk = bp + (size_t)k0 * ldb;
      v2f b0, b1;
      b0[0] = bk[bc0];           b0[1] = bk[(size_t)ldb + bc0];
      b1[0] = bk[bc1];           b1[1] = bk[(size_t)ldb + bc1];
      acc0 = __builtin_amdgcn_wmma_f32_16x16x4_f32(false, a, false, b0, (short)0, acc0, false, false);
      acc1 = __builtin_amdgcn_wmma_f32_16x16x4_f32(false, a, false, b1, (short)0, acc1, false, false);
    }
  }
  bool bok0 = bcol0 < N, bok1 = bcol1 < N;
  float bv0 = (bias && bok0) ? bias[bcol0] : 0.f;
  float bv1 = (bias && bok1) ? bias[bcol1] : 0.f;
#pragma unroll
  for (int r = 0; r < 8; ++r) {
    int m = tm * 16 + half * 8 + r;             // C/D: row = 8*half + vgpr
    if (m < M) {
      float* crow = C + (size_t)m * ldc;
      if (bok0) {
        float v = acc0[r] * alpha + bv0;
        if (accum) crow[bcol0] += v; else crow[bcol0] = v;
      }
      if (bok1) {
        float v = acc1[r] * alpha + bv1;
        if (accum) crow[bcol1] += v; else crow[bcol1] = v;
      }
    }
  }
}

// ---------------- LayerNorm(residual) ----------------
// out[r] = LN(x[r] + res[r]) * g + b   (rows of width K_E=384, 12 elems/lane)
__global__ void ln_residual(float* out, const float* x, const float* res,
                            const float* __restrict__ g, const float* __restrict__ b,
                            int rows)
{
  int wave = (blockIdx.x * blockDim.x + threadIdx.x) >> 5;
  int lane = threadIdx.x & 31;
  if (wave >= rows) return;
  const float* px = x + (size_t)wave * K_E;
  const float* pr = res ? res + (size_t)wave * K_E : nullptr;
  float v[12];
  float s = 0.f;
  for (int i = 0; i < 12; ++i) {
    int e = lane + i * 32;
    float t = px[e] + (pr ? pr[e] : 0.f);
    v[i] = t; s += t;
  }
  s = wave_sum(s);
  float mean = s * (1.f / K_E);
  float s2 = 0.f;
  for (int i = 0; i < 12; ++i) { float d = v[i] - mean; s2 += d * d; }
  s2 = wave_sum(s2);
  float inv = rsqrtf(s2 * (1.f / K_E) + 1e-5f);
  float* po = out + (size_t)wave * K_E;
  for (int i = 0; i < 12; ++i) {
    int e = lane + i * 32;
    po[e] = (v[i] - mean) * inv * g[e] + b[e];
  }
}

// ---------------- row softmax (in place) ----------------
__global__ void softmax_rows(float* S, int rows, int W)
{
  int wave = (blockIdx.x * blockDim.x + threadIdx.x) >> 5;
  int lane = threadIdx.x & 31;
  if (wave >= rows) return;
  float* p = S + (size_t)wave * W;
  float m = -1e30f;
  for (int j = lane; j < W; j += 32) m = fmaxf(m, p[j]);
  m = wave_max(m);
  float sum = 0.f;
  for (int j = lane; j < W; j += 32) { float e = expf(p[j] - m); p[j] = e; sum += e; }
  sum = wave_sum(sum);
  float inv = 1.f / sum;
  for (int j = lane; j < W; j += 32) p[j] *= inv;
}

// ---------------- exact GELU ----------------
__global__ void gelu_k(float* x, long n)
{
  long t = (long)blockIdx.x * blockDim.x + threadIdx.x;
  if (t >= n) return;
  float v = x[t];
  x[t] = 0.5f * v * (1.f + erff(v * 0.70710678118654752f));
}

// ---------------- seq init: emb[inputs] + rel_pos[min(l,7)] ----------------
__global__ void seq_init(float* seq, const int* __restrict__ inputs,
                         const float* __restrict__ emb, const float* __restrict__ rel_pos)
{
  int t = blockIdx.x * blockDim.x + threadIdx.x;
  if (t >= K_B * K_L * K_E) return;
  int e = t % K_E;
  int l = (t / K_E) % K_L;
  int b = t / (K_E * K_L);
  int tok = inputs[b * K_L + l];
  int pos = l < 7 ? l : 7;
  seq[t] = emb[(size_t)tok * K_E + e] + rel_pos[(size_t)pos * K_E + e];
}

// ---------------- pairs[p,t,:] = seq[b, i+t] + rel_pos[t] ----------------
__global__ void build_pairs(float* pairs, const float* __restrict__ seq,
                            const float* __restrict__ rel_pos, int Lc)
{
  int P = K_B * (Lc - 1);
  int t = blockIdx.x * blockDim.x + threadIdx.x;
  if (t >= P * 2 * K_E) return;
  int e = t % K_E;
  int tok = (t / K_E) & 1;
  int p = t / (2 * K_E);
  int b = p / (Lc - 1);
  int i = p % (Lc - 1);
  pairs[t] = seq[((size_t)(b * Lc + i + tok)) * K_E + e] + rel_pos[(size_t)tok * K_E + e];
}

// ---------------- conv_w (O,I,3) -> Wtap[k][O][I] ----------------
__global__ void repack_conv(float* wtap, const float* __restrict__ conv_w)
{
  int t = blockIdx.x * blockDim.x + threadIdx.x;
  if (t >= 3 * K_E * K_E) return;
  int k = t / (K_E * K_E);
  int o = (t / K_E) % K_E;
  int i = t % K_E;
  wtap[(size_t)k * K_E * K_E + (size_t)o * K_E + i] = conv_w[((size_t)o * K_E + i) * 3 + k];
}

// ---------------- 2-token MHA (HP=4, dh=96) ----------------
// thread = (p, h, d); qkv rows (2P x 1152): q|k|v, head h -> cols h*96..
__global__ void pair_attn(float* o, const float* __restrict__ qkv, int P)
{
  int t = blockIdx.x * blockDim.x + threadIdx.x;
  if (t >= P * K_HP * 96) return;
  int d = t % 96;
  int h = (t / 96) % K_HP;
  int p = t / (96 * K_HP);
  const float* q0 = qkv + (size_t)(2 * p) * 1152 + h * 96;
  const float* q1 = q0 + 1152;
  const float* k0 = q0 + 384; const float* k1 = k0 + 1152;
  const float* v0 = q0 + 768; const float* v1 = v0 + 1152;
  float s00 = 0, s01 = 0, s10 = 0, s11 = 0;
  for (int j = 0; j < 96; ++j) {
    float a0 = q0[j], a1 = q1[j], b0 = k0[j], b1 = k1[j];
    s00 += a0 * b0; s01 += a0 * b1; s10 += a1 * b0; s11 += a1 * b1;
  }
  const float sc = 0.10206207261596575f;  // 1/sqrt(96)
  s00 *= sc; s01 *= sc; s10 *= sc; s11 *= sc;
  float m0 = fmaxf(s00, s01), m1 = fmaxf(s10, s11);
  float e00 = expf(s00 - m0), e01 = expf(s01 - m0);
  float e10 = expf(s10 - m1), e11 = expf(s11 - m1);
  float i0 = 1.f / (e00 + e01), i1 = 1.f / (e10 + e11);
  float vd0 = v0[d], vd1 = v1[d];
  o[(size_t)(2 * p) * K_E + h * 96 + d]     = (e00 * vd0 + e01 * vd1) * i0;
  o[(size_t)(2 * p + 1) * K_E + h * 96 + d] = (e10 * vd0 + e11 * vd1) * i1;
}

// ---------------- sigmoid(hid[:, -1] . w + b): one wave per pair ----------------
__global__ void score_sigmoid(float* sc, const float* __restrict__ x,
                              const float* __restrict__ w, const float* __restrict__ bptr,
                              int P)
{
  int wave = (blockIdx.x * blockDim.x + threadIdx.x) >> 5;
  int lane = threadIdx.x & 31;
  if (wave >= P) return;
  const float* row = x + ((size_t)(2 * wave + 1)) * K_E;   // token index 1
  float s = 0.f;
  for (int i = 0; i < 12; ++i) { int e = lane + i * 32; s += row[e] * w[e]; }
  s = wave_sum(s);
  if (lane == 0) sc[wave] = 1.f / (1.f + expf(-(s + bptr[0])));
}

// ---------------- per-batch argmax + softmax entropy ----------------
__global__ void argmax_entropy(int* idx, float* loss, const float* __restrict__ sc,
                               int n, int iter)
{
  int b = blockIdx.x * blockDim.x + threadIdx.x;
  if (b >= K_B) return;
  const float* s = sc + (size_t)b * n;
  float m = s[0]; int am = 0;
  for (int i = 1; i < n; ++i) if (s[i] > m) { m = s[i]; am = i; }  // first max (jnp.argmax)
  idx[b] = am;
  float sum = 0.f;
  for (int i = 0; i < n; ++i) sum += expf(s[i] - m);
  float lse = m + logf(sum);
  float ent = 0.f;
  for (int i = 0; i < n; ++i) { float lp = s[i] - lse; ent -= expf(lp) * lp; }
  loss[(size_t)b * K_LOSS + iter] = ent;
}

__global__ void zero_loss_col(float* loss, int iter)
{
  int b = blockIdx.x * blockDim.x + threadIdx.x;
  if (b < K_B) loss[(size_t)b * K_LOSS + iter] = 0.f;
}

// ---------------- sel[b] = hid[b, idx[b]] ----------------
__global__ void gather_sel(float* sel, const float* __restrict__ x,
                           const int* __restrict__ idx, int Lc)
{
  int t = blockIdx.x * blockDim.x + threadIdx.x;
  if (t >= K_B * K_E) return;
  int e = t % K_E, b = t / K_E;
  int p = b * (Lc - 1) + idx[b];
  sel[t] = x[((size_t)(2 * p + 1)) * K_E + e];
}

// ---------------- memory attention (HM=6, dh=64, keys = 500 rel + nloc) ----
// block per (b, qi, h); q2/kloc/vloc rows: b*nq+qi / b*nloc+j
__global__ void mem_attn(float* o, const float* __restrict__ q2,
                         const float* __restrict__ krel, const float* __restrict__ vrel,
                         const float* __restrict__ kloc, const float* __restrict__ vloc,
                         int nq, int nloc)
{
  int h = blockIdx.x % K_HM;
  int bq = blockIdx.x / K_HM;
  int qi = bq % nq;
  int b = bq / nq;
  const int nk = K_R + nloc;
  __shared__ float sc[512];
  __shared__ float part[128];
  __shared__ float qs[64];
  int tid = threadIdx.x;
  const float* qrow = q2 + ((size_t)(b * nq + qi)) * K_E + h * 64;
  if (tid < 64) qs[tid] = qrow[tid];
  __syncthreads();
  for (int j = tid; j < nk; j += 128) {
    const float* krow = (j < K_R) ? (krel + (size_t)j * K_E + h * 64)
                                  : (kloc + ((size_t)(b * nloc + (j - K_R))) * K_E + h * 64);
    float s = 0.f;
    for (int d = 0; d < 64; ++d) s += qs[d] * krow[d];
    sc[j] = s * 0.125f;                                     // 1/sqrt(64)
  }
  __syncthreads();
  float m = -1e30f;
  for (int j = tid; j < nk; j += 128) m = fmaxf(m, sc[j]);
  part[tid] = m; __syncthreads();
  for (int s2 = 64; s2; s2 >>= 1) { if (tid < s2) part[tid] = fmaxf(part[tid], part[tid + s2]); __syncthreads(); }
  m = part[0]; __syncthreads();
  float sum = 0.f;
  for (int j = tid; j < nk; j += 128) { float e = expf(sc[j] - m); sc[j] = e; sum += e; }
  part[tid] = sum; __syncthreads();
  for (int s2 = 64; s2; s2 >>= 1) { if (tid < s2) part[tid] += part[tid + s2]; __syncthreads(); }
  float inv = 1.f / part[0];
  __syncthreads();
  if (tid < 64) {
    float acc = 0.f;
    for (int j = 0; j < nk; ++j) {
      const float* vrow = (j < K_R) ? (vrel + (size_t)j * K_E + h * 64)
                                    : (vloc + ((size_t)(b * nloc + (j - K_R))) * K_E + h * 64);
      acc += sc[j] * vrow[tid];
    }
    o[((size_t)(b * nq + qi)) * K_E + h * 64 + tid] = acc * inv;
  }
}

// ---------------- seq update: replace idx with fused, drop idx+1 ----------------
__global__ void update_seq(float* nseq, const float* __restrict__ seq,
                           const float* __restrict__ fused, const int* __restrict__ idx,
                           int Lc)
{
  int Ln = Lc - 1;
  int t = blockIdx.x * blockDim.x + threadIdx.x;
  if (t >= K_B * Ln * K_E) return;
  int e = t % K_E;
  int j = (t / K_E) % Ln;
  int b = t / (K_E * Ln);
  int id = idx[b];
  float v;
  if (j == id) v = fused[(size_t)b * K_E + e];
  else {
    int src = j + (j >= id + 1 ? 1 : 0);
    v = seq[((size_t)(b * Lc + src)) * K_E + e];
  }
  nseq[t] = v;
}

// ---------------- final logits + entropy + pred ----------------
// block per (b,t): logits[b,t,m] = x[b,t].mem[b,m] / sqrt(E); mem = [relmem(500); seq(2)]
__global__ void final_logits(float* pred, float* loss, const float* __restrict__ x,
                             const float* __restrict__ relmem, const float* __restrict__ seq)
{
  int t = blockIdx.x & 1;
  int b = blockIdx.x >> 1;
  __shared__ float lg[512];
  __shared__ float part[128];
  __shared__ float xs[K_E];
  int tid = threadIdx.x;
  const float* xr = x + ((size_t)(b * 2 + t)) * K_E;
  for (int e = tid; e < K_E; e += 128) xs[e] = xr[e];
  __syncthreads();
  const float invs = 0.051031036307982884f;   // 1/sqrt(384)
  for (int m = tid; m < K_PRED; m += 128) {
    const float* row = (m < K_R) ? (relmem + (size_t)m * K_E)
                                 : (seq + ((size_t)(b * 2 + (m - K_R))) * K_E);
    float s = 0.f;
    for (int e = 0; e < K_E; ++e) s += xs[e] * row[e];
    lg[m] = s * invs;
  }
  __syncthreads();
  if (t == 1) {
    for (int m = tid; m < K_PRED; m += 128) pred[(size_t)b * K_PRED + m] = lg[m];
  }
  float mm = -1e30f;
  for (int m = tid; m < K_PRED; m += 128) mm = fmaxf(mm, lg[m]);
  part[tid] = mm; __syncthreads();
  for (int s2 = 64; s2; s2 >>= 1) { if (tid < s2) part[tid] = fmaxf(part[tid], part[tid + s2]); __syncthreads(); }
  mm = part[0]; __syncthreads();
  float sum = 0.f;
  for (int m = tid; m < K_PRED; m += 128) sum += expf(lg[m] - mm);
  part[tid] = sum; __syncthreads();
  for (int s2 = 64; s2; s2 >>= 1) { if (tid < s2) part[tid] += part[tid + s2]; __syncthreads(); }
  float lse = mm + logf(part[0]);
  __syncthreads();
  float ent = 0.f;
  for (int m = tid; m < K_PRED; m += 128) { float lp = lg[m] - lse; ent -= expf(lp) * lp; }
  part[tid] = ent; __syncthreads();
  for (int s2 = 64; s2; s2 >>= 1) { if (tid < s2) part[tid] += part[tid + s2]; __syncthreads(); }
  if (tid == 0) loss[(size_t)b * K_LOSS + 7 + t] = part[0];
}

// ================= host orchestration =================
static inline int cdiv(int a, int b) { return (a + b - 1) / b; }

static void launch_gemm(hipStream_t s, bool bt, float* C, const float* A, const float* B,
                        const float* bias, int M, int N, int K, int lda, int ldb, int ldc,
                        float alpha, int accum)
{
  int npair = cdiv(cdiv(N, 16), 2);
  int tiles = cdiv(M, 16) * npair;               // each wave = 16x32 output block
  int blocks = cdiv(tiles, 4);                   // 4 waves per 128-thread block
  if (bt) gemm_wmma<true ><<<blocks, 128, 0, s>>>(C, A, B, bias, M, N, K, lda, ldb, ldc, alpha, accum);
  else    gemm_wmma<false><<<blocks, 128, 0, s>>>(C, A, B, bias, M, N, K, lda, ldb, ldc, alpha, accum);
}
static void launch_ln(hipStream_t s, float* out, const float* x, const float* res,
                      const float* g, const float* b, int rows)
{
  ln_residual<<<cdiv(rows, 4), 128, 0, s>>>(out, x, res, g, b, rows);
}

extern "C" void kernel_launch(void* const* d_in, const int* in_sizes, int n_in,
                              void* d_out, int out_size, void* d_ws, size_t ws_size,
                              hipStream_t stream)
{
  (void)in_sizes; (void)n_in; (void)out_size; (void)ws_size;
  // --- inputs (setup_inputs order) ---
  const int*   inputs     = (const int*)  d_in[0];
  const float* emb        = (const float*)d_in[1];
  const float* rel_pos    = (const float*)d_in[2];
  const float* qkv_dyn_w  = (const float*)d_in[3];
  const float* qkv_dyn_b  = (const float*)d_in[4];
  const float* out_dyn_w  = (const float*)d_in[5];
  const float* out_dyn_b  = (const float*)d_in[6];
  const float* ln_dyn_g   = (const float*)d_in[7];
  const float* ln_dyn_b   = (const float*)d_in[8];
  const float* conv_w     = (const float*)d_in[9];
  const float* conv_b     = (const float*)d_in[10];
  const float* tl_in_w    = (const float*)d_in[11];
  const float* tl_in_b    = (const float*)d_in[12];
  const float* tl_out_w   = (const float*)d_in[13];
  const float* tl_out_b   = (const float*)d_in[14];
  const float* tl_ln1_g   = (const float*)d_in[15];
  const float* tl_ln1_b   = (const float*)d_in[16];
  const float* tl_ln2_g   = (const float*)d_in[17];
  const float* tl_ln2_b   = (const float*)d_in[18];
  const float* tl_ff1_w   = (const float*)d_in[19];
  const float* tl_ff1_b   = (const float*)d_in[20];
  const float* tl_ff2_w   = (const float*)d_in[21];
  const float* tl_ff2_b   = (const float*)d_in[22];
  const float* fc_score_w = (const float*)d_in[23];
  const float* fc_score_b = (const float*)d_in[24];
  const float* fcq_w      = (const float*)d_in[25];
  const float* fcq_b      = (const float*)d_in[26];
  const float* fck_w      = (const float*)d_in[27];
  const float* fck_b      = (const float*)d_in[28];
  const float* fcv_w      = (const float*)d_in[29];
  const float* fcv_b      = (const float*)d_in[30];
  const float* mha_in_w   = (const float*)d_in[31];
  const float* mha_in_b   = (const float*)d_in[32];
  const float* mha_out_w  = (const float*)d_in[33];
  const float* mha_out_b  = (const float*)d_in[34];
  const float* lnf1_g     = (const float*)d_in[35];
  const float* lnf1_b     = (const float*)d_in[36];
  const float* lnf2_g     = (const float*)d_in[37];
  const float* lnf2_b     = (const float*)d_in[38];
  const float* ff1_w      = (const float*)d_in[39];
  const float* ff1_b      = (const float*)d_in[40];
  const float* ff2_w      = (const float*)d_in[41];
  const float* ff2_b      = (const float*)d_in[42];

  float* pred = (float*)d_out;                         // B x 502
  float* loss = (float*)d_out + (size_t)K_B * K_PRED;  // B x 9

  // --- workspace carve (deterministic) ---
  float* ws = (float*)d_ws;
  size_t off = 0;
  auto alloc = [&](size_t n) { float* p = ws + off; off += n; return p; };
  const int Pmax = K_B * (K_L - 1);                    // 896
  float* relqkv   = alloc((size_t)K_R * 1152);
  float* Smat     = alloc((size_t)2 * K_R * K_R);
  float* om       = alloc((size_t)K_R * K_E);
  float* om2      = alloc((size_t)K_R * K_E);
  float* relmem   = alloc((size_t)K_R * K_E);
  float* krel_lin = alloc((size_t)K_R * K_E);
  float* vrel_lin = alloc((size_t)K_R * K_E);
  float* krel2    = alloc((size_t)K_R * K_E);
  float* vrel2    = alloc((size_t)K_R * K_E);
  float* vrel2b   = alloc((size_t)K_R * K_E);          // final path: v-input = k_lin
  float* wtap     = alloc((size_t)3 * K_E * K_E);
  float* seqA     = alloc((size_t)K_B * K_L * K_E);
  float* seqB     = alloc((size_t)K_B * K_L * K_E);
  float* pairs    = alloc((size_t)Pmax * 2 * K_E);
  float* xp       = alloc((size_t)Pmax * 2 * K_E);
  float* qkv_p    = alloc((size_t)Pmax * 2 * 1152);
  float* o_p      = alloc((size_t)Pmax * 2 * K_E);
  float* h1_p     = alloc((size_t)Pmax * 2 * 1536);
  float* f_p      = alloc((size_t)Pmax * 2 * K_E);
  float* scores_p = alloc(Pmax);
  int*   idxbuf   = (int*)alloc(K_B);
  float* sel      = alloc((size_t)K_B * K_E);
  float* qlin     = alloc((size_t)2 * K_B * K_E);
  float* klin     = alloc((size_t)2 * K_B * K_E);
  float* vlin     = alloc((size_t)2 * K_B * K_E);
  float* q2       = alloc((size_t)2 * K_B * K_E);
  float* k2       = alloc((size_t)2 * K_B * K_E);
  float* v2       = alloc((size_t)2 * K_B * K_E);
  float* oatt     = alloc((size_t)2 * K_B * K_E);
  float* aatt     = alloc((size_t)2 * K_B * K_E);
  float* xf       = alloc((size_t)2 * K_B * K_E);
  float* h1f      = alloc((size_t)2 * K_B * 1536);
  float* ffb      = alloc((size_t)2 * K_B * K_E);
  float* fusedb   = alloc((size_t)K_B * K_E);

  const float sc192 = 0.07216878364870323f;            // 1/sqrt(192)

  // ===== precompute: relation memory (HD=2, dh=192) =====
  launch_gemm(stream, true, relqkv, emb, qkv_dyn_w, qkv_dyn_b,
              K_R, 1152, K_E, K_E, K_E, 1152, 1.f, 0);
  for (int h = 0; h < 2; ++h) {
    float* Sh = Smat + (size_t)h * K_R * K_R;
    launch_gemm(stream, true, Sh, relqkv + h * 192, relqkv + 384 + h * 192, nullptr,
                K_R, K_R, 192, 1152, 1152, K_R, sc192, 0);
  }
  softmax_rows<<<cdiv(2 * K_R, 4), 128, 0, stream>>>(Smat, 2 * K_R, K_R);
  for (int h = 0; h < 2; ++h) {
    float* Sh = Smat + (size_t)h * K_R * K_R;
    launch_gemm(stream, false, om + h * 192, Sh, relqkv + 768 + h * 192, nullptr,
                K_R, 192, K_R, K_R, 1152, K_E, 1.f, 0);
  }
  launch_gemm(stream, true, om2, om, out_dyn_w, out_dyn_b, K_R, K_E, K_E, K_E, K_E, K_E, 1.f, 0);
  launch_ln(stream, relmem, om2, emb, ln_dyn_g, ln_dyn_b, K_R);

  // batch-invariant projections of the relation half of `mem` (computed ONCE)
  launch_gemm(stream, true, krel_lin, relmem, fck_w, fck_b, K_R, K_E, K_E, K_E, K_E, K_E, 1.f, 0);
  launch_gemm(stream, true, vrel_lin, relmem, fcv_w, fcv_b, K_R, K_E, K_E, K_E, K_E, K_E, 1.f, 0);
  launch_gemm(stream, true, krel2, krel_lin, mha_in_w + 1 * K_E * K_E, mha_in_b + K_E,
              K_R, K_E, K_E, K_E, K_E, K_E, 1.f, 0);
  launch_gemm(stream, true, vrel2, vrel_lin, mha_in_w + 2 * K_E * K_E, mha_in_b + 2 * K_E,
              K_R, K_E, K_E, K_E, K_E, K_E, 1.f, 0);
  launch_gemm(stream, true, vrel2b, krel_lin, mha_in_w + 2 * K_E * K_E, mha_in_b + 2 * K_E,
              K_R, K_E, K_E, K_E, K_E, K_E, 1.f, 0);

  repack_conv<<<cdiv(3 * K_E * K_E, 256), 256, 0, stream>>>(wtap, conv_w);
  seq_init<<<cdiv(K_B * K_L * K_E, 256), 256, 0, stream>>>(seqA, inputs, emb, rel_pos);

  // ===== merge iterations =====
  float* seqcur = seqA;
  float* seqnxt = seqB;
  for (int it = 0; it < K_L - 1; ++it) {
    int Lc = K_L - it;                                 // 8,7,6,5,4,3,2
    if (Lc <= 2) {
      zero_loss_col<<<1, 128, 0, stream>>>(loss, it);
      continue;
    }
    int P = K_B * (Lc - 1);
    int rows2 = 2 * P;
    build_pairs<<<cdiv(P * 2 * K_E, 256), 256, 0, stream>>>(pairs, seqcur, rel_pos, Lc);

    // conv1d(k=3, pad=1) over length-2 sequences as 4 tap-GEMMs
    launch_gemm(stream, true, xp,        pairs,        wtap + 1 * K_E * K_E, conv_b,
                P, K_E, K_E, 2 * K_E, K_E, 2 * K_E, 1.f, 0);
    launch_gemm(stream, true, xp,        pairs + K_E,  wtap + 2 * K_E * K_E, nullptr,
                P, K_E, K_E, 2 * K_E, K_E, 2 * K_E, 1.f, 1);
    launch_gemm(stream, true, xp + K_E,  pairs,        wtap + 0 * K_E * K_E, conv_b,
                P, K_E, K_E, 2 * K_E, K_E, 2 * K_E, 1.f, 0);
    launch_gemm(stream, true, xp + K_E,  pairs + K_E,  wtap + 1 * K_E * K_E, nullptr,
                P, K_E, K_E, 2 * K_E, K_E, 2 * K_E, 1.f, 1);

    // two transformer layers
    for (int l = 0; l < 2; ++l) {
      launch_gemm(stream, true, qkv_p, xp, tl_in_w + (size_t)l * 3 * K_E * K_E,
                  tl_in_b + (size_t)l * 3 * K_E, rows2, 1152, K_E, K_E, K_E, 1152, 1.f, 0);
      pair_attn<<<cdiv(P * K_HP * 96, 256), 256, 0, stream>>>(o_p, qkv_p, P);
      launch_gemm(stream, true, f_p, o_p, tl_out_w + (size_t)l * K_E * K_E,
                  tl_out_b + (size_t)l * K_E, rows2, K_E, K_E, K_E, K_E, K_E, 1.f, 0);
      launch_ln(stream, xp, f_p, xp, tl_ln1_g + l * K_E, tl_ln1_b + l * K_E, rows2);
      launch_gemm(stream, true, h1_p, xp, tl_ff1_w + (size_t)l * 4 * K_E * K_E,
                  tl_ff1_b + (size_t)l * 4 * K_E, rows2, 1536, K_E, K_E, K_E, 1536, 1.f, 0);
      gelu_k<<<cdiv(rows2 * 1536, 256), 256, 0, stream>>>(h1_p, (long)rows2 * 1536);
      launch_gemm(stream, true, f_p, h1_p, tl_ff2_w + (size_t)l * K_E * 4 * K_E,
                  tl_ff2_b + (size_t)l * K_E, rows2, K_E, 1536, 1536, 1536, K_E, 1.f, 0);
      launch_ln(stream, xp, f_p, xp, tl_ln2_g + l * K_E, tl_ln2_b + l * K_E, rows2);
    }

    // score / argmax / entropy / select
    score_sigmoid<<<cdiv(P, 4), 128, 0, stream>>>(scores_p, xp, fc_score_w, fc_score_b, P);
    argmax_entropy<<<1, 128, 0, stream>>>(idxbuf, loss, scores_p, Lc - 1, it);
    gather_sel<<<cdiv(K_B * K_E, 256), 256, 0, stream>>>(sel, xp, idxbuf, Lc);

    // fuse (relation half of k/v precomputed -> only the selected row per batch here)
    launch_gemm(stream, true, qlin, sel, fcq_w, fcq_b, K_B, K_E, K_E, K_E, K_E, K_E, 1.f, 0);
    launch_gemm(stream, true, klin, sel, fck_w, fck_b, K_B, K_E, K_E, K_E, K_E, K_E, 1.f, 0);
    launch_gemm(stream, true, vlin, sel, fcv_w, fcv_b, K_B, K_E, K_E, K_E, K_E, K_E, 1.f, 0);
    launch_gemm(stream, true, q2, qlin, mha_in_w, mha_in_b, K_B, K_E, K_E, K_E, K_E, K_E, 1.f, 0);
    launch_gemm(stream, true, k2, klin, mha_in_w + 1 * K_E * K_E, mha_in_b + K_E,
                K_B, K_E, K_E, K_E, K_E, K_E, 1.f, 0);
    launch_gemm(stream, true, v2, vlin, mha_in_w + 2 * K_E * K_E, mha_in_b + 2 * K_E,
                K_B, K_E, K_E, K_E, K_E, K_E, 1.f, 0);
    mem_attn<<<K_B * 1 * K_HM, 128, 0, stream>>>(oatt, q2, krel2, vrel2, k2, v2, 1, 1);
    launch_gemm(stream, true, aatt, oatt, mha_out_w, mha_out_b, K_B, K_E, K_E, K_E, K_E, K_E, 1.f, 0);
    launch_ln(stream, xf, aatt, sel, lnf1_g, lnf1_b, K_B);
    launch_gemm(stream, true, h1f, xf, ff1_w, ff1_b, K_B, 1536, K_E, K_E, K_E, 1536, 1.f, 0);
    gelu_k<<<cdiv(K_B * 1536, 256), 256, 0, stream>>>(h1f, (long)K_B * 1536);
    launch_gemm(stream, true, ffb, h1f, ff2_w, ff2_b, K_B, K_E, 1536, 1536, 1536, K_E, 1.f, 0);
    launch_ln(stream, fusedb, ffb, xf, lnf2_g, lnf2_b, K_B);

    update_seq<<<cdiv(K_B * (Lc - 1) * K_E, 256), 256, 0, stream>>>(seqnxt, seqcur, fusedb,
                                                                    idxbuf, Lc);
    float* tmp = seqcur; seqcur = seqnxt; seqnxt = tmp;
  }

  // ===== final block (seq length 2; note v-input = k here) =====
  const int rowsF = 2 * K_B;
  launch_gemm(stream, true, qlin, seqcur, fcq_w, fcq_b, rowsF, K_E, K_E, K_E, K_E, K_E, 1.f, 0);
  launch_gemm(stream, true, klin, seqcur, fck_w, fck_b, rowsF, K_E, K_E, K_E, K_E, K_E, 1.f, 0);
  launch_gemm(stream, true, q2, qlin, mha_in_w, mha_in_b, rowsF, K_E, K_E, K_E, K_E, K_E, 1.f, 0);
  launch_gemm(stream, true, k2, klin, mha_in_w + 1 * K_E * K_E, mha_in_b + K_E,
              rowsF, K_E, K_E, K_E, K_E, K_E, 1.f, 0);
  launch_gemm(stream, true, v2, klin, mha_in_w + 2 * K_E * K_E, mha_in_b + 2 * K_E,
              rowsF, K_E, K_E, K_E, K_E, K_E, 1.f, 0);
  mem_attn<<<K_B * 2 * K_HM, 128, 0, stream>>>(oatt, q2, krel2, vrel2b, k2, v2, 2, 2);
  launch_gemm(stream, true, aatt, oatt, mha_out_w, mha_out_b, rowsF, K_E, K_E, K_E, K_E, K_E, 1.f, 0);
  launch_ln(stream, xf, aatt, seqcur, lnf1_g, lnf1_b, rowsF);
  launch_gemm(stream, true, h1f, xf, ff1_w, ff1_b, rowsF, 1536, K_E, K_E, K_E, 1536, 1.f, 0);
  gelu_k<<<cdiv(rowsF * 1536, 256), 256, 0, stream>>>(h1f, (long)rowsF * 1536);
  launch_gemm(stream, true, ffb, h1f, ff2_w, ff2_b, rowsF, K_E, 1536, 1536, 1536, K_E, 1.f, 0);
  launch_ln(stream, xf, ffb, xf, lnf2_g, lnf2_b, rowsF);

  final_logits<<<K_B * 2, 128, 0, stream>>>(pred, loss, xf, relmem, seqcur);
}